// YOLOLayer_56014963474784
// MI455X (gfx1250) — compile-verified
//
#include <hip/hip_runtime.h>
#include <hip/hip_bf16.h>

// ---------------------------------------------------------------------------
// YOLO layer: decode (sigmoid/exp + anchor scale) -> corner boxes via WMMA
// -> bitonic sort by score desc (stable, index-asc tiebreak) -> greedy NMS
// (single wave, lane-parallel IoU) -> masked writeout.
// Target: gfx1250 (CDNA5, wave32). WMMA: V_WMMA_F32_16X16X4_F32 does the
// exact-f32 center->corner linear transform for 16 boxes per wave.
// ---------------------------------------------------------------------------

typedef __attribute__((ext_vector_type(2))) float v2f;
typedef __attribute__((ext_vector_type(8))) float v8f;

#define NBOX   12288      // 3 anchors * 64 * 64
#define NSORT  16384      // next pow2
#define FSIZE  64
#define STRIDE_F 8.0f
#define CONF_THRE 0.01f
#define NMS_THRE  0.45f

__device__ __forceinline__ float sigmoidf_(float x) {
    return 1.0f / (1.0f + __expf(-x));
}

// ---------------------------------------------------------------------------
// Kernel 1: decode. One wave handles a tile of 16 boxes; WMMA produces the
// 16x4 corner matrix (x1,y1,x2,y2) from the 16x4 center matrix (x,y,w,h).
// Grid: 96 blocks x 256 threads (8 waves/block, 768 tiles total) -> all lanes
// active, EXEC == all-ones at the WMMA (ISA requirement).
// ---------------------------------------------------------------------------
__global__ void yolo_decode_kernel(const float* __restrict__ feat,
                                   const float* __restrict__ anchors,
                                   float* __restrict__ boxes,   // [NBOX][4] xyxy
                                   float* __restrict__ obj,     // [NBOX]
                                   float* __restrict__ cls,     // [NBOX]
                                   float* __restrict__ score)   // [NBOX]
{
    const int tid  = threadIdx.x;
    const int wave = tid >> 5;
    const int lane = tid & 31;
    const int l16  = lane & 15;
    const bool hi  = lane >= 16;

    const int tile = blockIdx.x * 8 + wave;   // 16-box tile id, 0..767
    const int n    = tile * 16 + l16;         // this lane's box (dup in hi half)

    const int a   = n >> 12;                  // anchor index 0..2
    const int rem = n & 4095;                 // gy*64+gx
    const int gy  = rem >> 6;
    const int gx  = rem & 63;

    const float* fb = feat + (a * 6) * 4096 + rem;
    __builtin_prefetch(fb, 0, 1);             // global_prefetch_b8

    const float f0 = fb[0 * 4096];
    const float f1 = fb[1 * 4096];
    const float f2 = fb[2 * 4096];
    const float f3 = fb[3 * 4096];
    const float f4 = fb[4 * 4096];
    const float f5 = fb[5 * 4096];

    const float aw = anchors[2 * a + 0];      // (anchor/STRIDE)*STRIDE == anchor
    const float ah = anchors[2 * a + 1];

    const float x = (sigmoidf_(f0) + (float)gx) * STRIDE_F;
    const float y = (sigmoidf_(f1) + (float)gy) * STRIDE_F;
    const float w = __expf(f2) * aw;
    const float h = __expf(f3) * ah;
    const float so = sigmoidf_(f4);
    const float sc = sigmoidf_(f5);

    if (!hi) {
        obj[n]   = so;
        cls[n]   = sc;
        score[n] = so * sc;
    }

    // A-matrix 16x4 f32 (ISA layout): lane L   -> row L, K=0 (VGPR0), K=1 (VGPR1)
    //                                 lane16+L -> row L, K=2 (VGPR0), K=3 (VGPR1)
    v2f A;
    A.x = hi ? w : x;
    A.y = hi ? h : y;

    // B-matrix 4x16: col0=x1, col1=y1, col2=x2, col3=y2 transform; cols 4..15 = 0
    //   B[0][n]: 1 at n=0,2   B[1][n]: 1 at n=1,3
    //   B[2][n]: -0.5 at n=0, +0.5 at n=2   B[3][n]: -0.5 at n=1, +0.5 at n=3
    v2f B;
    if (!hi) {
        B.x = (l16 == 0 || l16 == 2) ? 1.0f : 0.0f;   // K=0 row
        B.y = (l16 == 1 || l16 == 3) ? 1.0f : 0.0f;   // K=1 row
    } else {
        B.x = (l16 == 0) ? -0.5f : (l16 == 2) ? 0.5f : 0.0f;  // K=2 row
        B.y = (l16 == 1) ? -0.5f : (l16 == 3) ? 0.5f : 0.0f;  // K=3 row
    }

    v8f C = {};
    // (neg_a, A, neg_b, B, c_mod, C, reuse_a, reuse_b)
    v8f D = __builtin_amdgcn_wmma_f32_16x16x4_f32(
        false, A, false, B, (short)0, C, false, false);

    // D 16x16 f32 layout: VGPR r, lanes 0-15 -> M=r,N=lane ; lanes16-31 -> M=8+r
    // Lane l16<4 (lo half) holds column l16 of rows 0..7; hi half rows 8..15.
    if (l16 < 4) {
        const int col  = l16;
        const int base = tile * 16 + (hi ? 8 : 0);
#pragma unroll
        for (int k = 0; k < 8; ++k) {
            boxes[(base + k) * 4 + col] = D[k];
        }
    }
}

// ---------------------------------------------------------------------------
// Kernel 2: build sort keys. key = (~orderable(score)) << 32 | index.
// Ascending u64 sort => score descending, index ascending on ties (matches
// stable jnp.argsort(-scores)). Pad slots get ~0ull (sort to end).
// ---------------------------------------------------------------------------
__global__ void yolo_build_keys_kernel(const float* __restrict__ score,
                                       unsigned long long* __restrict__ keys)
{
    const int n = blockIdx.x * blockDim.x + threadIdx.x;
    if (n >= NSORT) return;
    if (n < NBOX) {
        unsigned u = __float_as_uint(score[n]);
        unsigned m = (u & 0x80000000u) ? ~u : (u | 0x80000000u); // ascending map
        unsigned key = ~m;                                       // descending
        keys[n] = ((unsigned long long)key << 32) | (unsigned)n;
    } else {
        keys[n] = ~0ull;
    }
}

// ---------------------------------------------------------------------------
// Kernel 3: one bitonic compare-exchange step (ascending overall).
// ---------------------------------------------------------------------------
__global__ void yolo_bitonic_step_kernel(unsigned long long* __restrict__ keys,
                                         unsigned j, unsigned k)
{
    const unsigned i = blockIdx.x * blockDim.x + threadIdx.x;
    if (i >= NSORT) return;
    const unsigned ixj = i ^ j;
    if (ixj > i) {
        const unsigned long long a = keys[i];
        const unsigned long long b = keys[ixj];
        const bool up = ((i & k) == 0);
        if ((a > b) == up) {
            keys[i]   = b;
            keys[ixj] = a;
        }
    }
}

// ---------------------------------------------------------------------------
// Kernel 4: greedy NMS, one wave. Sequential over sorted candidates; the 32
// lanes stride over the kept list computing IoU; __any() reduces. Lane 0
// appends kept boxes to the global list (+fence for cross-lane visibility).
// ---------------------------------------------------------------------------
__global__ void yolo_nms_kernel(const unsigned long long* __restrict__ keys,
                                const float* __restrict__ boxes,
                                const float* __restrict__ score,
                                int* __restrict__ keep,
                                float* __restrict__ kept)
{
    const int lane = threadIdx.x;
    int kcnt = 0;

    for (int i = 0; i < NSORT; ++i) {
        const unsigned long long p = keys[i];
        const unsigned idx = (unsigned)(p & 0xFFFFFFFFull);
        if (idx >= (unsigned)NBOX) continue;   // padding

        const float sc = score[idx];
        if (!(sc >= CONF_THRE)) {
            if (lane == 0) keep[idx] = 0;
            continue;
        }

        const float bx1 = boxes[idx * 4 + 0];
        const float by1 = boxes[idx * 4 + 1];
        const float bx2 = boxes[idx * 4 + 2];
        const float by2 = boxes[idx * 4 + 3];
        const float area = (bx2 - bx1) * (by2 - by1);

        int sup = 0;
        for (int t = lane; t < kcnt; t += 32) {
            const float kx1 = kept[t * 4 + 0];
            const float ky1 = kept[t * 4 + 1];
            const float kx2 = kept[t * 4 + 2];
            const float ky2 = kept[t * 4 + 3];
            const float tlx = fmaxf(bx1, kx1);
            const float tly = fmaxf(by1, ky1);
            const float brx = fminf(bx2, kx2);
            const float bry = fminf(by2, ky2);
            float inter = (brx - tlx) * (bry - tly);
            inter = (tlx < brx && tly < bry) ? inter : 0.0f;
            const float karea = (kx2 - kx1) * (ky2 - ky1);
            const float iou = inter / (area + karea - inter);
            if (iou >= NMS_THRE) sup = 1;
        }
        sup = __any(sup);

        if (!sup) {
            if (lane == 0) {
                kept[kcnt * 4 + 0] = bx1;
                kept[kcnt * 4 + 1] = by1;
                kept[kcnt * 4 + 2] = bx2;
                kept[kcnt * 4 + 3] = by2;
                keep[idx] = 1;
            }
            kcnt++;
            __threadfence();   // make lane0's append visible to all lanes
        } else {
            if (lane == 0) keep[idx] = 0;
        }
    }
}

// ---------------------------------------------------------------------------
// Kernel 5: masked writeout: det = [x1,y1,x2,y2,obj,cls] * keep ; keep as f32.
// ---------------------------------------------------------------------------
__global__ void yolo_finalize_kernel(const float* __restrict__ boxes,
                                     const float* __restrict__ obj,
                                     const float* __restrict__ cls,
                                     const int* __restrict__ keep,
                                     float* __restrict__ out)
{
    const int n = blockIdx.x * blockDim.x + threadIdx.x;
    if (n >= NBOX) return;
    const float k = keep[n] ? 1.0f : 0.0f;
    out[n * 6 + 0] = boxes[n * 4 + 0] * k;
    out[n * 6 + 1] = boxes[n * 4 + 1] * k;
    out[n * 6 + 2] = boxes[n * 4 + 2] * k;
    out[n * 6 + 3] = boxes[n * 4 + 3] * k;
    out[n * 6 + 4] = obj[n] * k;
    out[n * 6 + 5] = cls[n] * k;
    out[NBOX * 6 + n] = k;
}

// ---------------------------------------------------------------------------
extern "C" void kernel_launch(void* const* d_in, const int* in_sizes, int n_in,
                              void* d_out, int out_size, void* d_ws, size_t ws_size,
                              hipStream_t stream)
{
    (void)in_sizes; (void)n_in; (void)out_size; (void)ws_size;

    const float* feat    = (const float*)d_in[0];   // (1, 18, 64, 64) f32
    const float* anchors = (const float*)d_in[1];   // (3, 2) f32
    float* out = (float*)d_out;                     // 12288*6 det + 12288 keep

    // Workspace layout (bytes):
    //   0       boxes  [NBOX][4] f32  (196608)
    //   196608  obj    [NBOX]    f32  ( 49152)
    //   245760  cls    [NBOX]    f32  ( 49152)
    //   294912  score  [NBOX]    f32  ( 49152)
    //   344064  keep   [NBOX]    i32  ( 49152)
    //   393216  keys   [NSORT]   u64  (131072)
    //   524288  kept   [NBOX][4] f32  (196608)
    char* ws = (char*)d_ws;
    float* boxes = (float*)(ws + 0);
    float* obj   = (float*)(ws + 196608);
    float* cls   = (float*)(ws + 245760);
    float* score = (float*)(ws + 294912);
    int*   keep  = (int*)  (ws + 344064);
    unsigned long long* keys = (unsigned long long*)(ws + 393216);
    float* kept  = (float*)(ws + 524288);

    // 1) decode + WMMA corner transform: 768 tiles of 16 boxes, 8 waves/block
    yolo_decode_kernel<<<96, 256, 0, stream>>>(feat, anchors, boxes, obj, cls, score);

    // 2) sort keys
    yolo_build_keys_kernel<<<NSORT / 256, 256, 0, stream>>>(score, keys);

    // 3) bitonic sort (ascending u64 == score desc, index asc)
    for (unsigned k = 2; k <= NSORT; k <<= 1) {
        for (unsigned j = k >> 1; j >= 1; j >>= 1) {
            yolo_bitonic_step_kernel<<<NSORT / 256, 256, 0, stream>>>(keys, j, k);
        }
    }

    // 4) greedy NMS (single wave32)
    yolo_nms_kernel<<<1, 32, 0, stream>>>(keys, boxes, score, keep, kept);

    // 5) masked writeout
    yolo_finalize_kernel<<<NBOX / 256, 256, 0, stream>>>(boxes, obj, cls, keep, out);
}